// ModelNew_14061722927803
// MI455X (gfx1250) — compile-verified
//
#include <hip/hip_runtime.h>
#include <math.h>

// ---------------------------------------------------------------------------
// Conv3x3(8->64) + GroupNorm(16) + scale + maxpool4 + clamp, for MI455X.
// Two passes, recompute conv in pass 2 (x is L2-resident: 67MB of 192MB L2)
// instead of spilling the 520MB conv intermediate to HBM.
// Conv is implicit GEMM: M=16 spatial, N=64 out-ch, K=72 via fp32 WMMA
// V_WMMA_F32_16X16X4_F32 (18 K-steps), full fp32 precision.
//
// K-ordering: k = kw*24 + (ci*3 + kh).  x tile is staged TRANSPOSED in LDS
// as xs[col][q] (q = ci*3+kh, row stride 28 dwords): the two A operands of
// each K-step are adjacent dwords -> one aligned ds_load_b64 straight into
// a consecutive VGPR pair, per-step offset kw*28+q0 is a ds immediate, and
// 28j mod 64 is all-distinct -> bank-conflict-free. B pair-rows padded to
// 80 float2 (=160 dwords, 32 mod 64) so lo/hi half-wave hit disjoint banks.
// ---------------------------------------------------------------------------

typedef __attribute__((ext_vector_type(2))) float v2f;
typedef __attribute__((ext_vector_type(8))) float v8f;

#define CI       8
#define HIN      128
#define WIN      128
#define CO       64
#define HOUT     126
#define WOUT     126
#define QROWS    24            // rows per conv window: 8 ci * 3 kh
#define QSTRIDE  28            // padded q-stride (bank-conflict-free)
#define XCOLS    130           // 128 data cols + 2 zero-pad cols
#define XWIN     (XCOLS * QSTRIDE)   // floats per staged window (3640)
#define BSTRIDE  80            // wps pair-row stride in float2 (bank padding)
#define POOLED   31
#define GCNT     (4 * 126 * 126)     // elements per (n, group)

// Weights into LDS as K-pairs in the permuted K order:
// wps[p*BSTRIDE + co] = (w[perm(2p)], w[perm(2p+1)]),
// perm(k) = ci*9 + kh*3 + kw with ci=(k%24)/3, kh=(k%24)%3, kw=k/24.
// f32 WMMA A/B register layout: VGPR0 = {K0 | K2}, VGPR1 = {K1 | K3}.
__device__ __forceinline__ void load_weights(const float* __restrict__ cw,
                                             float2* wps, int tid) {
    for (int idx = tid; idx < 36 * 64; idx += 256) {
        const int p  = idx % 36;
        const int co = idx / 36;
        const int k0 = 2 * p;
        const int k1 = 2 * p + 1;
        const int o0 = ((k0 % 24) / 3) * 9 + ((k0 % 24) % 3) * 3 + (k0 / 24);
        const int o1 = ((k1 % 24) / 3) * 9 + ((k1 % 24) % 3) * 3 + (k1 / 24);
        wps[p * BSTRIDE + co] = make_float2(cw[co * 72 + o0], cw[co * 72 + o1]);
    }
}

// Stage one 24-row conv window transposed: xw[col*QSTRIDE + q] = x[ci][row+kh][col].
// Pure-shift indexing (row is exactly 128 wide); cols 128/129 zero-padded.
__device__ __forceinline__ void stage_window(const float* __restrict__ x,
                                             float* xw, int n, int row0, int tid) {
    for (int idx = tid; idx < QROWS * 128; idx += 256) {
        const int q   = idx >> 7;       // 0..23 = ci*3 + kh
        const int col = idx & 127;
        const int ci  = q / 3;
        const int rr  = q % 3;
        xw[col * QSTRIDE + q] = x[((n * CI + ci) * HIN + (row0 + rr)) * WIN + col];
    }
    if (tid < 2 * QROWS)                // zero pad cols 128,129
        xw[(128 + (tid & 1)) * QSTRIDE + (tid >> 1)] = 0.0f;
}

// One 16(spatial) x 64(co) conv tile: 18 unrolled K=4 steps x 4 co-tiles.
// xsA  = xw + wm*QSTRIDE + 2*hi   (per-lane, fixed for the whole tile)
// wpsB = wps + hi*BSTRIDE + lane15
// All per-step offsets are compile-time constants (ds immediates).
__device__ __forceinline__ void conv_tile(const float* xsA, const float2* wpsB,
                                          v8f c[4]) {
#pragma unroll
    for (int p = 0; p < 18; ++p) {
        const int k0 = 4 * p;
        const int q0 = k0 % 24;     // row within window
        const int kw = k0 / 24;     // column shift
        const v2f a = *(const v2f*)(xsA + kw * QSTRIDE + q0);  // K=k0,k0+1 (+2 hi)
#pragma unroll
        for (int ct = 0; ct < 4; ++ct) {
            const v2f b = *(const v2f*)(wpsB + (k0 / 2) * BSTRIDE + ct * 16);
            c[ct] = __builtin_amdgcn_wmma_f32_16x16x4_f32(
                false, a, false, b, (short)0, c[ct], false, false);
        }
    }
}

// Pass 1: conv + bias, accumulate per-(n,group) sum / sumsq. Block = (h, n).
__global__ __launch_bounds__(256) void conv_stats_kernel(
    const float* __restrict__ x, const float* __restrict__ cw,
    const float* __restrict__ cb, float* __restrict__ stats) {
    __shared__ __align__(16) float  xs[XWIN];
    __shared__ __align__(16) float2 wps[36 * BSTRIDE];
    __shared__ float sstats[32];

    const int tid = threadIdx.x;
    const int h   = blockIdx.x;
    const int n   = blockIdx.y;

    load_weights(cw, wps, tid);
    if (tid < 32) sstats[tid] = 0.0f;
    stage_window(x, xs, n, h, tid);
    __syncthreads();

    const int wid    = tid >> 5;
    const int lane   = tid & 31;
    const int lane15 = lane & 15;
    const int hi     = lane >> 4;
    const int w0     = wid * 16;

    const float*  xsA  = xs + (w0 + lane15) * QSTRIDE + 2 * hi;
    const float2* wpsB = wps + hi * BSTRIDE + lane15;

    v8f c[4] = {};
    conv_tile(xsA, wpsB, c);

    // C layout: reg r, lane -> y[w = w0 + r + 8*hi][co = ct*16 + lane15]
    const int wBase = w0 + 8 * hi;
#pragma unroll
    for (int ct = 0; ct < 4; ++ct) {
        const int   co   = ct * 16 + lane15;
        const float bias = cb[co];
        float s = 0.0f, s2 = 0.0f;
#pragma unroll
        for (int r = 0; r < 8; ++r) {
            if (wBase + r < WOUT) {   // mask tail of tile 7 (w >= 126)
                const float v = c[ct][r] + bias;
                s  += v;
                s2 += v * v;
            }
        }
        // lanes {4q..4q+3} u {16+4q..16+4q+3} share one group -> xor 1,2,16
        s  += __shfl_xor(s, 1, 32);  s  += __shfl_xor(s, 2, 32);  s  += __shfl_xor(s, 16, 32);
        s2 += __shfl_xor(s2, 1, 32); s2 += __shfl_xor(s2, 2, 32); s2 += __shfl_xor(s2, 16, 32);
        if ((lane & 0x13) == 0) {
            const int g = co >> 2;
            atomicAdd(&sstats[g * 2 + 0], s);
            atomicAdd(&sstats[g * 2 + 1], s2);
        }
    }
    __syncthreads();
    if (tid < 32) atomicAdd(&stats[n * 32 + tid], sstats[tid]);
}

// Pass 2: recompute conv, fused affine (GN+scale+bias), 4x4 maxpool, clamp.
// Block = (ph, n): 4 conv rows -> one pooled row [64co][31pw].
// x staged as 4 transposed windows (one per hh).
__global__ __launch_bounds__(256) void conv_pool_kernel(
    const float* __restrict__ x, const float* __restrict__ cw,
    const float* __restrict__ cb, const float* __restrict__ gw,
    const float* __restrict__ gb, const float* __restrict__ sc,
    const float* __restrict__ stats, float* __restrict__ out) {
    __shared__ __align__(16) float  xs[4 * XWIN];     // ~58 KB
    __shared__ __align__(16) float2 wps[36 * BSTRIDE];
    __shared__ float amul_s[64];
    __shared__ float badd_s[64];
    __shared__ float sOut[64 * 32];

    const int tid = threadIdx.x;
    const int ph  = blockIdx.x;
    const int n   = blockIdx.y;

    load_weights(cw, wps, tid);
    if (tid < 64) {
        const int   co   = tid;
        const int   g    = co >> 2;
        const float s    = stats[n * 32 + g * 2 + 0];
        const float s2   = stats[n * 32 + g * 2 + 1];
        const float mean = s / (float)GCNT;
        const float var  = s2 / (float)GCNT - mean * mean;
        const float rinv = rsqrtf(var + 1e-5f);
        amul_s[co] = rinv * gw[co] * sc[co];
        badd_s[co] = ((cb[co] - mean) * rinv * gw[co] + gb[co]) * sc[co];
    }
    for (int hh = 0; hh < 4; ++hh)
        stage_window(x, xs + hh * XWIN, n, 4 * ph + hh, tid);
    __syncthreads();

    const int wid    = tid >> 5;
    const int lane   = tid & 31;
    const int lane15 = lane & 15;
    const int hi     = lane >> 4;
    const int w0     = wid * 16;

    const float*  xsA0 = xs + (w0 + lane15) * QSTRIDE + 2 * hi;
    const float2* wpsB = wps + hi * BSTRIDE + lane15;

    float pm[4][2];
#pragma unroll
    for (int ct = 0; ct < 4; ++ct) pm[ct][0] = pm[ct][1] = -__builtin_inff();

    for (int hh = 0; hh < 4; ++hh) {
        v8f c[4] = {};
        conv_tile(xsA0 + hh * XWIN, wpsB, c);
#pragma unroll
        for (int ct = 0; ct < 4; ++ct) {
            const int   co = ct * 16 + lane15;
            const float a  = amul_s[co];
            const float bb = badd_s[co];
#pragma unroll
            for (int r = 0; r < 8; ++r) {
                const float v = c[ct][r] * a + bb;
                pm[ct][r >> 2] = fmaxf(pm[ct][r >> 2], v);
            }
        }
    }
    // lane element (ct,j) -> pooled[co = ct*16+lane15][pw = 4*wid + 2*hi + j]
#pragma unroll
    for (int ct = 0; ct < 4; ++ct) {
        const int co = ct * 16 + lane15;
#pragma unroll
        for (int j = 0; j < 2; ++j) {
            const float v = fminf(fmaxf(pm[ct][j], 0.0f), 1.0f);
            sOut[co * 32 + (4 * wid + 2 * hi + j)] = v;  // col 31 = dead pad
        }
    }
    __syncthreads();
    for (int idx = tid; idx < 64 * POOLED; idx += 256) {
        const int co = idx / POOLED;
        const int pw = idx % POOLED;
        out[((n * CO + co) * POOLED + ph) * POOLED + pw] = sOut[co * 32 + pw];
    }
}

extern "C" void kernel_launch(void* const* d_in, const int* in_sizes, int n_in,
                              void* d_out, int out_size, void* d_ws, size_t ws_size,
                              hipStream_t stream) {
    const float* x  = (const float*)d_in[0];
    const float* cw = (const float*)d_in[1];
    const float* cb = (const float*)d_in[2];
    const float* gw = (const float*)d_in[3];
    const float* gb = (const float*)d_in[4];
    const float* sc = (const float*)d_in[5];
    float*       out   = (float*)d_out;
    float*       stats = (float*)d_ws;  // [128][16 groups][2] = 16 KB

    hipMemsetAsync(stats, 0, 128 * 32 * sizeof(float), stream);
    conv_stats_kernel<<<dim3(HOUT, 128), 256, 0, stream>>>(x, cw, cb, stats);
    conv_pool_kernel<<<dim3(POOLED, 128), 256, 0, stream>>>(x, cw, cb, gw, gb,
                                                            sc, stats, out);
}